// TopTPercentAggregationFunction_58119497449641
// MI455X (gfx1250) — compile-verified
//
#include <hip/hip_runtime.h>

// Top-2%-mean per (b,c) row of cam[32,4,512,512] f32.
// Exact 32-bit radix-select (12+12+8 bit counting passes) to find the k-th
// largest value's exact bit pattern, then one deterministic sum pass.
// Memory-bound: 4 x 128MB streams. CDNA5 paths: async global->LDS staging
// (global_load_async_to_lds_b128 + s_wait_asynccnt) and LDS atomics.

constexpr int ROWS    = 128;        // 32*4
constexpr int HW      = 512 * 512;  // 262144 elements per row
constexpr int TOP_T   = 5243;       // round(0.02 * 262144)
constexpr int THREADS = 512;        // 16 wave32 waves per workgroup

// Order-preserving map: larger float <-> larger unsigned key.
__device__ __forceinline__ unsigned key_of(float f) {
  unsigned u = __float_as_uint(f);
  return (u & 0x80000000u) ? ~u : (u | 0x80000000u);
}
__device__ __forceinline__ float val_of(unsigned k) {
  unsigned u = (k & 0x80000000u) ? (k & 0x7fffffffu) : ~k;
  return __uint_as_float(u);
}

// Issue a per-lane 16B async copy global -> LDS (VGLOBAL GV mode, ASYNCcnt).
__device__ __forceinline__ void async_tile_load_b128(unsigned lds_byte_addr,
                                                     const float* gptr) {
  unsigned long long ga = (unsigned long long)gptr;
  asm volatile("global_load_async_to_lds_b128 %0, %1, off"
               :: "v"(lds_byte_addr), "v"(ga)
               : "memory");
}

// Descending scan over a histogram: find first bin (from the top) where the
// cumulative count reaches `need`. Returns that bin and the count strictly above.
__device__ __forceinline__ void scan_desc(const unsigned* cnt, int nbins,
                                          unsigned need, unsigned* bin_out,
                                          unsigned* above_out) {
  unsigned cum = 0;
  int b = nbins - 1;
  for (; b > 0; --b) {
    if (cum + cnt[b] >= need) break;
    cum += cnt[b];
  }
  *bin_out = (unsigned)b;
  *above_out = cum;
}

// ---------------- Pass 1: histogram of key[31:20] (4096 bins) ----------------
// Data is staged through LDS with double-buffered async b128 loads.
__global__ __launch_bounds__(THREADS) void topt_pass1_hist12(
    const float* __restrict__ cam, unsigned* __restrict__ ws) {
  __shared__ unsigned cnt[4096];
  __shared__ float tile[2][THREADS * 4];

  const int tid = threadIdx.x;
  const int row = blockIdx.x;
  for (int i = tid; i < 4096; i += THREADS) cnt[i] = 0u;
  __syncthreads();

  const float* rowp = cam + (size_t)row * HW;
  constexpr int ITER = HW / (THREADS * 4);  // 128 iterations, 16B/lane/iter

  // Prologue: stage iteration 0 into buffer 0.
  async_tile_load_b128((unsigned)(unsigned long long)(void*)&tile[0][tid * 4],
                       rowp + (size_t)tid * 4);

  for (int it = 0; it < ITER; ++it) {
    const int nb = it + 1;
    if (nb < ITER) {
      async_tile_load_b128(
          (unsigned)(unsigned long long)(void*)&tile[nb & 1][tid * 4],
          rowp + ((size_t)nb * THREADS + tid) * 4);
      // Two ops in flight; async ops complete in order, so cnt<=1 means the
      // older (current iteration's) staging copy has landed in LDS.
      asm volatile("s_wait_asynccnt 1" ::: "memory");
    } else {
      asm volatile("s_wait_asynccnt 0" ::: "memory");
    }
    float4 v = *(const float4*)(&tile[it & 1][tid * 4]);
    float f[4] = {v.x, v.y, v.z, v.w};
#pragma unroll
    for (int j = 0; j < 4; ++j) {
      unsigned k = key_of(f[j]);
      atomicAdd(&cnt[k >> 20], 1u);  // ds_add_u32
    }
  }
  __syncthreads();

  if (tid == 0) {
    unsigned B1, above;
    scan_desc(cnt, 4096, TOP_T, &B1, &above);
    ws[row * 8 + 0] = B1;
    ws[row * 8 + 1] = above;
  }
}

// ------------- Pass 2: within bin B1, histogram of key[19:8] ----------------
__global__ __launch_bounds__(THREADS) void topt_pass2_hist12(
    const float* __restrict__ cam, unsigned* __restrict__ ws) {
  __shared__ unsigned cnt[4096];
  const int tid = threadIdx.x;
  const int row = blockIdx.x;
  const unsigned B1   = ws[row * 8 + 0];
  const unsigned cnt1 = ws[row * 8 + 1];
  for (int i = tid; i < 4096; i += THREADS) cnt[i] = 0u;
  __syncthreads();

  const float4* rowp = (const float4*)(cam + (size_t)row * HW);
  for (int i = tid; i < HW / 4; i += THREADS) {
    float4 v = rowp[i];
    float f[4] = {v.x, v.y, v.z, v.w};
#pragma unroll
    for (int j = 0; j < 4; ++j) {
      unsigned k = key_of(f[j]);
      if ((k >> 20) == B1) atomicAdd(&cnt[(k >> 8) & 0xfffu], 1u);
    }
  }
  __syncthreads();

  if (tid == 0) {
    unsigned need = (unsigned)TOP_T - cnt1;  // >= 1 by construction
    unsigned B2, above;
    scan_desc(cnt, 4096, need, &B2, &above);
    ws[row * 8 + 2] = B2;
    ws[row * 8 + 3] = above;
  }
}

// ------ Pass 3: within 24-bit prefix, histogram of key[7:0] -> exact K ------
__global__ __launch_bounds__(THREADS) void topt_pass3_hist8(
    const float* __restrict__ cam, unsigned* __restrict__ ws) {
  __shared__ unsigned cnt[256];
  const int tid = threadIdx.x;
  const int row = blockIdx.x;
  const unsigned B1   = ws[row * 8 + 0];
  const unsigned cnt1 = ws[row * 8 + 1];
  const unsigned B2   = ws[row * 8 + 2];
  const unsigned cnt2 = ws[row * 8 + 3];
  const unsigned prefix = (B1 << 12) | B2;  // key[31:8]
  for (int i = tid; i < 256; i += THREADS) cnt[i] = 0u;
  __syncthreads();

  const float4* rowp = (const float4*)(cam + (size_t)row * HW);
  for (int i = tid; i < HW / 4; i += THREADS) {
    float4 v = rowp[i];
    float f[4] = {v.x, v.y, v.z, v.w};
#pragma unroll
    for (int j = 0; j < 4; ++j) {
      unsigned k = key_of(f[j]);
      if ((k >> 8) == prefix) atomicAdd(&cnt[k & 0xffu], 1u);
    }
  }
  __syncthreads();

  if (tid == 0) {
    unsigned need = (unsigned)TOP_T - cnt1 - cnt2;  // >= 1
    unsigned B3, above;
    scan_desc(cnt, 256, need, &B3, &above);
    unsigned K = (prefix << 8) | B3;      // exact bit pattern of k-th largest
    unsigned remaining = need - above;    // #tied elements (value == val_of(K))
    ws[row * 8 + 4] = K;
    ws[row * 8 + 5] = remaining;
  }
}

// --------- Pass 4: deterministic sum of values with key > K, finalize -------
__global__ __launch_bounds__(THREADS) void topt_pass4_sum(
    const float* __restrict__ cam, const unsigned* __restrict__ ws,
    float* __restrict__ out) {
  __shared__ float red[THREADS];
  const int tid = threadIdx.x;
  const int row = blockIdx.x;
  const unsigned K = ws[row * 8 + 4];
  const unsigned remaining = ws[row * 8 + 5];

  const float4* rowp = (const float4*)(cam + (size_t)row * HW);
  float s = 0.0f;
  for (int i = tid; i < HW / 4; i += THREADS) {
    float4 v = rowp[i];
    float f[4] = {v.x, v.y, v.z, v.w};
#pragma unroll
    for (int j = 0; j < 4; ++j) {
      if (key_of(f[j]) > K) s += f[j];
    }
  }
  red[tid] = s;
  __syncthreads();
  for (int off = THREADS / 2; off > 0; off >>= 1) {
    if (tid < off) red[tid] += red[tid + off];
    __syncthreads();
  }
  if (tid == 0) {
    out[row] = (red[0] + (float)remaining * val_of(K)) / (float)TOP_T;
  }
}

extern "C" void kernel_launch(void* const* d_in, const int* in_sizes, int n_in,
                              void* d_out, int out_size, void* d_ws,
                              size_t ws_size, hipStream_t stream) {
  const float* cam = (const float*)d_in[0];
  float* out = (float*)d_out;
  unsigned* ws = (unsigned*)d_ws;  // 128 rows * 8 u32 = 4 KB used

  dim3 grid(ROWS), block(THREADS);
  topt_pass1_hist12<<<grid, block, 0, stream>>>(cam, ws);
  topt_pass2_hist12<<<grid, block, 0, stream>>>(cam, ws);
  topt_pass3_hist8 <<<grid, block, 0, stream>>>(cam, ws);
  topt_pass4_sum   <<<grid, block, 0, stream>>>(cam, ws, out);
}